// GATLinkPredictor_31748398252251
// MI455X (gfx1250) — compile-verified
//
#include <hip/hip_runtime.h>
#include <math.h>

// ---------------------------------------------------------------------------
// GAT (2-layer, PyG GATConv v1) for MI455X / gfx1250.
//   Layer 1: H=4 heads, C=32, concat -> 128, relu(out + b1)
//   Layer 2: H=1 head,  C=128, mean(=identity), out + b2
// GEMMs use V_WMMA_F32_16X16X4_F32 (exact fp32 math, wave32 matrix pipe).
// Edge softmax/aggregation uses L2-resident atomics (working set ~55MB << 192MB L2).
// ---------------------------------------------------------------------------

typedef float v2f __attribute__((ext_vector_type(2)));
typedef float v8f __attribute__((ext_vector_type(8)));

#define NEG_SLOPE 0.2f
// enc(-inf): bits(-inf)=0xFF800000, sign set -> ~bits = 0x007FFFFF
#define ENC_NEG_INF 0x007FFFFFu

__device__ __forceinline__ float leaky_relu(float x) {
  return x > 0.f ? x : NEG_SLOPE * x;
}
// order-preserving float -> uint mapping (for atomicMax on floats incl. negatives)
__device__ __forceinline__ unsigned enc_f32(float f) {
  unsigned u = __float_as_uint(f);
  return (u & 0x80000000u) ? ~u : (u | 0x80000000u);
}
__device__ __forceinline__ float dec_f32(unsigned u) {
  return (u & 0x80000000u) ? __uint_as_float(u & 0x7FFFFFFFu)
                           : __uint_as_float(~u);
}

// ---------------------------------------------------------------------------
__global__ void fill_u32(unsigned* __restrict__ p, unsigned v, long long n) {
  long long i = (long long)blockIdx.x * blockDim.x + threadIdx.x;
  long long stride = (long long)gridDim.x * blockDim.x;
  for (; i < n; i += stride) p[i] = v;
}

// ---------------------------------------------------------------------------
// C[M,N] = A[M,K] @ B[K,N], all fp32 row-major. One wave per 16x16 C tile,
// K-loop in steps of 4 via V_WMMA_F32_16X16X4_F32.
__global__ void gemm_wmma_f32(const float* __restrict__ A,
                              const float* __restrict__ B,
                              float* __restrict__ C, int M, int N, int K) {
  const int lane = threadIdx.x & 31;
  const int wave = threadIdx.x >> 5;
  const int wavesPerBlock = blockDim.x >> 5;
  const int tilesN = N >> 4;
  const int tile = blockIdx.x * wavesPerBlock + wave;
  const int tm = tile / tilesN;
  const int tn = tile - tm * tilesN;
  if (tm * 16 >= M) return;

  const int hl = lane >> 4;    // half-wave: 0 or 1
  const int l16 = lane & 15;
  int arow = tm * 16 + l16;
  if (arow >= M) arow = M - 1;             // clamp (tile fully in-range for our M)
  const int bcol = tn * 16 + l16;
  const int koff = hl * 2;                 // K sub-offset for this half-wave

  const float* __restrict__ ap = A + (long long)arow * K + koff;
  v8f acc = {};

  for (int k = 0; k < K; k += 4) {
    // A 16x4 tile: lane holds A[arow][k+koff], A[arow][k+koff+1] (contiguous)
    v2f a;
    a.x = ap[k + 0];
    a.y = ap[k + 1];
    // B 4x16 tile: lane holds B[k+koff][bcol], B[k+koff+1][bcol]
    const float* __restrict__ bp = B + (long long)(k + koff) * N + bcol;
    v2f b;
    b.x = bp[0];
    b.y = bp[N];
    acc = __builtin_amdgcn_wmma_f32_16x16x4_f32(
        /*neg_a=*/false, a, /*neg_b=*/false, b,
        /*c_mod=*/(short)0, acc, /*reuse_a=*/false, /*reuse_b=*/false);
  }

  // D layout: VGPR r, lane l -> D[tm*16 + r + 8*hl][bcol]
#pragma unroll
  for (int r = 0; r < 8; ++r) {
    const int orow = tm * 16 + r + 8 * hl;
    if (orow < M) C[(long long)orow * N + bcol] = acc[r];
  }
}

// ---------------------------------------------------------------------------
// al_src[n,h] = sum_c h[n,h,c]*a_src[h,c]; al_dst likewise. One thread per (n,h).
__global__ void attn_logits(const float* __restrict__ h,
                            const float* __restrict__ a_src,
                            const float* __restrict__ a_dst,
                            float* __restrict__ alsrc, float* __restrict__ aldst,
                            int Nn, int H, int C) {
  const int HC = H * C;
  long long idx = (long long)blockIdx.x * blockDim.x + threadIdx.x;
  const long long total = (long long)Nn * H;
  const long long stride = (long long)gridDim.x * blockDim.x;
  for (; idx < total; idx += stride) {
    const int n = (int)(idx / H);
    const int hh = (int)(idx - (long long)n * H);
    const float* __restrict__ hp = h + (long long)n * HC + hh * C;
    const float* __restrict__ as = a_src + hh * C;
    const float* __restrict__ ad = a_dst + hh * C;
    float ss = 0.f, sd = 0.f;
    for (int c = 0; c < C; ++c) {
      const float v = hp[c];
      ss += v * as[c];
      sd += v * ad[c];
    }
    alsrc[idx] = ss;
    aldst[idx] = sd;
  }
}

// ---------------------------------------------------------------------------
__device__ __forceinline__ void edge_endpoints(const int* __restrict__ ei,
                                               long long e, int E, int* src,
                                               int* dst) {
  if (e < E) {
    *src = ei[e];
    *dst = ei[(long long)E + e];
  } else {  // self loops appended
    *src = *dst = (int)(e - E);
  }
}

// scatter-max of leaky(al_src[src]+al_dst[dst]) into m[dst,h] (encoded uint)
__global__ void edge_max(const int* __restrict__ ei, int E, int Nn, int H,
                         const float* __restrict__ alsrc,
                         const float* __restrict__ aldst,
                         unsigned* __restrict__ m) {
  const long long Etot = (long long)E + Nn;
  long long idx = (long long)blockIdx.x * blockDim.x + threadIdx.x;
  const long long stride = (long long)gridDim.x * blockDim.x;
  for (; idx < Etot; idx += stride) {
    int src, dst;
    edge_endpoints(ei, idx, E, &src, &dst);
    for (int hh = 0; hh < H; ++hh) {
      const float e = leaky_relu(alsrc[(long long)src * H + hh] +
                                 aldst[(long long)dst * H + hh]);
      atomicMax(&m[(long long)dst * H + hh], enc_f32(e));
    }
  }
}

// scatter-add of exp(e - m[dst]) into s[dst,h]
__global__ void edge_expsum(const int* __restrict__ ei, int E, int Nn, int H,
                            const float* __restrict__ alsrc,
                            const float* __restrict__ aldst,
                            const unsigned* __restrict__ m,
                            float* __restrict__ s) {
  const long long Etot = (long long)E + Nn;
  long long idx = (long long)blockIdx.x * blockDim.x + threadIdx.x;
  const long long stride = (long long)gridDim.x * blockDim.x;
  for (; idx < Etot; idx += stride) {
    int src, dst;
    edge_endpoints(ei, idx, E, &src, &dst);
    for (int hh = 0; hh < H; ++hh) {
      const float e = leaky_relu(alsrc[(long long)src * H + hh] +
                                 aldst[(long long)dst * H + hh]);
      const float ex = __expf(e - dec_f32(m[(long long)dst * H + hh]));
      atomicAdd(&s[(long long)dst * H + hh], ex);
    }
  }
}

// wave-per-edge: agg[dst, :] += alpha[h(k)] * h[src, :]   (HC = 128 channels)
__global__ void edge_aggregate(const int* __restrict__ ei, int E, int Nn, int H,
                               int C, const float* __restrict__ alsrc,
                               const float* __restrict__ aldst,
                               const unsigned* __restrict__ m,
                               const float* __restrict__ s,
                               const float* __restrict__ h,
                               float* __restrict__ agg) {
  const int lane = threadIdx.x & 31;
  const long long wid =
      ((long long)blockIdx.x * blockDim.x + threadIdx.x) >> 5;
  const long long nwaves = ((long long)gridDim.x * blockDim.x) >> 5;
  const long long Etot = (long long)E + Nn;
  const int HC = H * C;
  for (long long e = wid; e < Etot; e += nwaves) {
    int src, dst;
    edge_endpoints(ei, e, E, &src, &dst);
    for (int k = lane; k < HC; k += 32) {
      const int hh = k / C;  // uniform across the wave for C>=32
      const float lg = leaky_relu(alsrc[(long long)src * H + hh] +
                                  aldst[(long long)dst * H + hh]);
      const float alpha = __expf(lg - dec_f32(m[(long long)dst * H + hh])) /
                          (s[(long long)dst * H + hh] + 1e-16f);
      atomicAdd(&agg[(long long)dst * HC + k],
                alpha * h[(long long)src * HC + k]);
    }
  }
}

// ---------------------------------------------------------------------------
__global__ void bias_act(const float* __restrict__ in,
                         const float* __restrict__ b, float* __restrict__ out,
                         long long total, int F, int do_relu) {
  long long i = (long long)blockIdx.x * blockDim.x + threadIdx.x;
  const long long stride = (long long)gridDim.x * blockDim.x;
  for (; i < total; i += stride) {
    float v = in[i] + b[(int)(i % F)];
    out[i] = do_relu ? fmaxf(v, 0.f) : v;
  }
}

// ---------------------------------------------------------------------------
static inline int cdiv(long long a, long long b) { return (int)((a + b - 1) / b); }

extern "C" void kernel_launch(void* const* d_in, const int* in_sizes, int n_in,
                              void* d_out, int out_size, void* d_ws,
                              size_t ws_size, hipStream_t stream) {
  const float* x  = (const float*)d_in[0];
  const int*   ei = (const int*)d_in[1];   // edge_index [2,E] (int32)
  const float* W1 = (const float*)d_in[2];
  const float* a1s = (const float*)d_in[3];
  const float* a1d = (const float*)d_in[4];
  const float* b1 = (const float*)d_in[5];
  const float* W2 = (const float*)d_in[6];
  const float* a2s = (const float*)d_in[7];
  const float* a2d = (const float*)d_in[8];
  const float* b2 = (const float*)d_in[9];
  float* z = (float*)d_out;

  const int F1 = in_sizes[5];        // 128 = H1*C1
  const int H1 = 4;
  const int C1 = F1 / H1;            // 32
  const int IN = in_sizes[2] / F1;   // 256
  const int Nn = in_sizes[0] / IN;   // 50000
  const int E  = in_sizes[1] / 2;    // 1.6M
  const int F2 = in_sizes[9];        // 128

  // workspace carve-up (floats): h | agg | alsrc | aldst | m | s  (~55 MB)
  float* h     = (float*)d_ws;
  float* agg   = h + (size_t)Nn * F1;
  float* alsrc = agg + (size_t)Nn * F1;
  float* aldst = alsrc + (size_t)Nn * H1;
  unsigned* m  = (unsigned*)(aldst + (size_t)Nn * H1);
  float* s     = (float*)(m + (size_t)Nn * H1);

  const long long Etot = (long long)E + Nn;
  const int TB = 256;

  // ---------------- Layer 1 ----------------
  {
    const int tiles = cdiv(Nn, 16) * (F1 / 16);
    gemm_wmma_f32<<<cdiv(tiles, TB / 32), TB, 0, stream>>>(x, W1, h, Nn, F1, IN);

    attn_logits<<<cdiv((long long)Nn * H1, TB), TB, 0, stream>>>(
        h, a1s, a1d, alsrc, aldst, Nn, H1, C1);

    fill_u32<<<cdiv((long long)Nn * F1, TB), TB, 0, stream>>>(
        (unsigned*)agg, 0u, (long long)Nn * F1);
    fill_u32<<<cdiv((long long)Nn * H1, TB), TB, 0, stream>>>(
        m, ENC_NEG_INF, (long long)Nn * H1);
    fill_u32<<<cdiv((long long)Nn * H1, TB), TB, 0, stream>>>(
        (unsigned*)s, 0u, (long long)Nn * H1);

    edge_max<<<cdiv(Etot, TB), TB, 0, stream>>>(ei, E, Nn, H1, alsrc, aldst, m);
    edge_expsum<<<cdiv(Etot, TB), TB, 0, stream>>>(ei, E, Nn, H1, alsrc, aldst,
                                                   m, s);
    edge_aggregate<<<cdiv(Etot * 32, TB), TB, 0, stream>>>(
        ei, E, Nn, H1, C1, alsrc, aldst, m, s, h, agg);

    // act1 = relu(agg + b1), in place
    bias_act<<<cdiv((long long)Nn * F1, TB), TB, 0, stream>>>(
        agg, b1, agg, (long long)Nn * F1, F1, 1);
  }

  // ---------------- Layer 2 (H=1) ----------------
  {
    const int H2 = 1, C2 = F2;
    const int tiles = cdiv(Nn, 16) * (F2 / 16);
    gemm_wmma_f32<<<cdiv(tiles, TB / 32), TB, 0, stream>>>(agg, W2, h, Nn, F2,
                                                           F1);

    attn_logits<<<cdiv((long long)Nn * H2, TB), TB, 0, stream>>>(
        h, a2s, a2d, alsrc, aldst, Nn, H2, C2);

    fill_u32<<<cdiv((long long)Nn * F2, TB), TB, 0, stream>>>(
        (unsigned*)agg, 0u, (long long)Nn * F2);
    fill_u32<<<cdiv((long long)Nn * H2, TB), TB, 0, stream>>>(
        m, ENC_NEG_INF, (long long)Nn * H2);
    fill_u32<<<cdiv((long long)Nn * H2, TB), TB, 0, stream>>>(
        (unsigned*)s, 0u, (long long)Nn * H2);

    edge_max<<<cdiv(Etot, TB), TB, 0, stream>>>(ei, E, Nn, H2, alsrc, aldst, m);
    edge_expsum<<<cdiv(Etot, TB), TB, 0, stream>>>(ei, E, Nn, H2, alsrc, aldst,
                                                   m, s);
    edge_aggregate<<<cdiv(Etot * 32, TB), TB, 0, stream>>>(
        ei, E, Nn, H2, C2, alsrc, aldst, m, s, h, agg);

    bias_act<<<cdiv((long long)Nn * F2, TB), TB, 0, stream>>>(
        agg, b2, z, (long long)Nn * F2, F2, 0);
  }
}